// DTWFeatures_c_14259291422993
// MI455X (gfx1250) — compile-verified
//
#include <hip/hip_runtime.h>

// Problem constants (match reference)
#define B_ 64
#define N_ 64
#define D_ 6
#define L_ 32
#define T_ 2048

#define CHUNK 256              // time steps staged per LDS buffer (1 KB)
#define NCHUNK (T_ / CHUNK)    // 8

typedef float v2f __attribute__((ext_vector_type(2)));

// One block per (b, d) row of x; 64 threads = 64 patterns (2 waves, wave32).
// Each lane runs the full open-end DTW DP for its (b, n, d) triple with the
// 32-entry DP column and 32-entry pattern row held in VGPRs.
// x[b,d,:] is staged into LDS in double-buffered 1KB chunks using the CDNA5
// async global->LDS copy (ASYNCcnt) so the copy overlaps the DP compute.
// Costs are computed with packed fp32 (v_pk_add/mul_f32); the serial (min,+)
// chain is reduced to exactly 2 dependent VALU ops per DP cell.
__global__ __launch_bounds__(64, 1)
void dtw_features_kernel(const float* __restrict__ x,
                         const float* __restrict__ patts,
                         const float* __restrict__ wp,
                         float* __restrict__ out)
{
    __shared__ __align__(16) float xs[2][CHUNK];

    const int tid = threadIdx.x;        // pattern index n
    const int row = blockIdx.x;         // b*D + d
    const float w = wp[0];

    const float* xrow = x + (size_t)row * T_;          // x[b][d][*] contiguous
    const int b = row / D_;
    const int d = row - b * D_;
    // out[b][n][d][*] : base offset ((b*N + n)*D + d) * T
    float* orow = out + ((size_t)b * N_ * D_ + (size_t)tid * D_ + (size_t)d) * T_;

    // ---- load pattern row into registers as fp32 pairs (128B per lane) ----
    v2f p2[L_ / 2];
    {
        const float4* p4 = (const float4*)(patts + tid * L_);
        #pragma unroll
        for (int i = 0; i < L_ / 4; ++i) {
            float4 v = p4[i];
            p2[2 * i + 0] = v2f{v.x, v.y};
            p2[2 * i + 1] = v2f{v.z, v.w};
        }
    }

    // DP column, previous time step; +inf == "no path yet" (free start handled
    // by the recurrence since w*inf stays inf until row 0 seeds a finite value)
    float Dc[L_];
    #pragma unroll
    for (int i = 0; i < L_; ++i) Dc[i] = __builtin_inff();

    // ---- CDNA5 async global->LDS prefetch: lane copies its 16B slice ----
    auto prefetch = [&](int c, int buf) {
        const void* g = (const void*)(xrow + c * CHUNK + tid * 4);
        // low 32 bits of a generic LDS pointer are the LDS byte address
        unsigned lds = (unsigned)(unsigned long long)(const void*)&xs[buf][tid * 4];
        asm volatile("global_load_async_to_lds_b128 %0, %1, off"
                     :: "v"(lds), "v"(g)
                     : "memory");
    };

    prefetch(0, 0);

    for (int c = 0; c < NCHUNK; ++c) {
        // wait for this wave's async copies, then sync both waves of the block
        asm volatile("s_wait_asynccnt 0" ::: "memory");
        __syncthreads();
        // issue next chunk's copy into the other buffer; safe because every
        // thread finished reading that buffer before reaching the barrier
        if (c + 1 < NCHUNK) prefetch(c + 1, (c + 1) & 1);

        const float4* xb = (const float4*)&xs[c & 1][0];
        float* op = orow + c * CHUNK;

        for (int q = 0; q < CHUNK / 4; ++q) {
            float4 xv4 = xb[q];                 // ds_load_b128 broadcast
            float xv_arr[4] = {xv4.x, xv4.y, xv4.z, xv4.w};
            float res[4];
            #pragma unroll
            for (int s = 0; s < 4; ++s) {
                const float xv = xv_arr[s];
                const v2f xv2 = {xv, xv};

                // Pass 1: packed cost  cc[i] = (p[i] - xv)^2
                // (v_pk_add_f32 with neg + v_pk_mul_f32: 32 ops for 32 cells)
                float cc[L_];
                #pragma unroll
                for (int k = 0; k < L_ / 2; ++k) {
                    v2f t = p2[k] - xv2;
                    v2f c2 = t * t;
                    cc[2 * k + 0] = c2.x;
                    cc[2 * k + 1] = c2.y;
                }

                // Pass 2: independent horizontal/diagonal mins of the OLD
                // column (all off the critical path, VOPD-pairable)
                float pm[L_];
                #pragma unroll
                for (int i = 1; i < L_; ++i)
                    pm[i] = fminf(Dc[i], Dc[i - 1]);

                // Pass 3: serial chain — 2 dependent ops per cell
                float prev = cc[0];             // D[0, j] = c0 (free start)
                Dc[0] = prev;
                #pragma unroll
                for (int i = 1; i < L_; ++i) {
                    float m = fminf(pm[i], prev);   // chain op 1
                    prev = fmaf(w, m, cc[i]);       // chain op 2
                    Dc[i] = prev;
                }
                res[s] = __builtin_sqrtf(prev); // output row L-1
            }
            float4 o4 = {res[0], res[1], res[2], res[3]};
            ((float4*)op)[q] = o4;              // 16B streaming store per lane
        }
    }
}

extern "C" void kernel_launch(void* const* d_in, const int* in_sizes, int n_in,
                              void* d_out, int out_size, void* d_ws, size_t ws_size,
                              hipStream_t stream) {
    const float* x     = (const float*)d_in[0];   // (B, D, T) fp32
    const float* patts = (const float*)d_in[1];   // (N, L)   fp32
    const float* w     = (const float*)d_in[2];   // scalar   fp32
    float* out = (float*)d_out;                   // (B, N, D, T) fp32

    dim3 grid(B_ * D_);   // 384 blocks: one per (b, d)
    dim3 block(N_);       // 64 threads = 2 wave32 waves, thread = pattern n
    hipLaunchKernelGGL(dtw_features_kernel, grid, block, 0, stream,
                       x, patts, w, out);
}